// SuperDecLoss_63848983822929
// MI455X (gfx1250) — compile-verified
//
#include <hip/hip_runtime.h>
#include <hip/hip_bf16.h>
#include <math.h>

// Problem constants (from reference setup_inputs)
#define BB 4
#define PP 24
#define NN 3072
#define SS 128
#define BP (BB * PP)          // 96
#define NTILES (NN / 16)      // 192
#define STILES (SS / 16)      // 8
#define TILES_PER_WAVE (NTILES / 8)  // 24
#define CUB_BLOCKS ((BB * NN * PP) / 256)  // 1152

typedef __attribute__((ext_vector_type(2))) float v2f;
typedef __attribute__((ext_vector_type(8))) float v8f;

__device__ __forceinline__ float fexp_f(float v, float e) {
    // sign(v) * |v|^e
    return copysignf(powf(fabsf(v), e), v);
}

__device__ __forceinline__ float fix_f(float v) {
    return (v > 0.f ? 1.f : -1.f) * fmaxf(fabsf(v), 1e-6f);
}

// ---------------------------------------------------------------------------
// Kernel 1: superquadric sample points X[b,p,s,3] and |X|^2
// ---------------------------------------------------------------------------
__global__ __launch_bounds__(256)
void sample_kernel(const float* __restrict__ scale, const float* __restrict__ shape,
                   const float* __restrict__ etas, const float* __restrict__ omegas,
                   float* __restrict__ X, float* __restrict__ x2) {
    int idx = blockIdx.x * blockDim.x + threadIdx.x;   // over BP*SS = 12288
    if (idx >= BP * SS) return;
    int bp = idx / SS;
    float eta = etas[idx];
    float omg = omegas[idx];
    if (eta == 0.f) eta = 1e-6f;
    if (omg == 0.f) omg = 1e-6f;
    float a1 = scale[bp * 3 + 0], a2 = scale[bp * 3 + 1], a3 = scale[bp * 3 + 2];
    float e1 = shape[bp * 2 + 0], e2 = shape[bp * 2 + 1];
    float ce = cosf(eta), se = sinf(eta);
    float co = cosf(omg), so = sinf(omg);
    float fce = fexp_f(ce, e1);
    float x = fix_f(a1 * fce * fexp_f(co, e2));
    float y = fix_f(a2 * fce * fexp_f(so, e2));
    float z = fix_f(a3 * fexp_f(se, e1));
    X[idx * 3 + 0] = x;
    X[idx * 3 + 1] = y;
    X[idx * 3 + 2] = z;
    x2[idx] = x * x + y * y + z * z;
}

// ---------------------------------------------------------------------------
// Kernel 2: WMMA Chamfer core. One workgroup per (b,p), 8 waves.
//   D[s][n] = x2[s] + p2[n] - 2 * (X[s] . pc[n])   -> cross term via
//   v_wmma_f32_16x16x4_f32 (K=3 padded to 4).
// Outputs: pclpart[bp] = sum_n min_s(D) * assign[b,n,p]
//          prim2[bp]   = mean_s min_n(D)
// ---------------------------------------------------------------------------
__global__ __launch_bounds__(256)
void wmma_chamfer_kernel(const float* __restrict__ pc_inver,
                         const float* __restrict__ am,
                         const float* __restrict__ X,
                         const float* __restrict__ x2,
                         float* __restrict__ pclpart,
                         float* __restrict__ prim2) {
    const int bp = blockIdx.x;
    const int b = bp / PP;
    const int p = bp % PP;
    const float* __restrict__ pc = pc_inver + (size_t)bp * NN * 3;
    const float* __restrict__ Xp = X + (size_t)bp * SS * 3;
    const float* __restrict__ x2p = x2 + (size_t)bp * SS;
    const float* __restrict__ amp = am + (size_t)b * NN * PP + p;  // stride PP over n

    const int lane = threadIdx.x & 31;
    const int wave = threadIdx.x >> 5;
    const bool hi = lane >= 16;
    const int col = lane & 15;

    __shared__ float lds_x2[SS];
    __shared__ float lds_prim[8 * SS];   // per-wave per-s min over that wave's n range
    __shared__ float lds_red[256];

    if (threadIdx.x < SS) lds_x2[threadIdx.x] = x2p[threadIdx.x];
    __syncthreads();

    // Preload A matrices for all 8 S-tiles.
    // A layout (16x4 f32): lanes 0-15: K0,K1 of row M=lane; lanes 16-31: K2,K3 of row M=lane-16.
    v2f a[STILES];
#pragma unroll
    for (int k = 0; k < STILES; ++k) {
        int m = 16 * k + col;
        if (!hi) {
            a[k].x = Xp[m * 3 + 0];
            a[k].y = Xp[m * 3 + 1];
        } else {
            a[k].x = Xp[m * 3 + 2];
            a[k].y = 0.f;   // K=3 pad
        }
    }

    float nmin[STILES * 8];
#pragma unroll
    for (int i = 0; i < STILES * 8; ++i) nmin[i] = 3.402823466e+38f;
    float pclsum = 0.f;

    for (int it = 0; it < TILES_PER_WAVE; ++it) {
        const int t = wave * TILES_PER_WAVE + it;
        const int n = t * 16 + col;
        // Prefetch next tile's pc data (global_prefetch_b8).
        if (it + 1 < TILES_PER_WAVE)
            __builtin_prefetch(&pc[(n + 16) * 3], 0, 1);
        float px = pc[n * 3 + 0];
        float py = pc[n * 3 + 1];
        float pz = pc[n * 3 + 2];
        v2f bm;
        bm.x = hi ? pz : px;
        bm.y = hi ? 0.f : py;
        float p2 = px * px + py * py + pz * pz;

        float colmin = 3.402823466e+38f;
#pragma unroll
        for (int k = 0; k < STILES; ++k) {
            v8f c = {};
            v8f d = __builtin_amdgcn_wmma_f32_16x16x4_f32(
                false, a[k], false, bm, (short)0, c, false, false);
#pragma unroll
            for (int r = 0; r < 8; ++r) {
                float sx2 = lds_x2[16 * k + r + (hi ? 8 : 0)];
                float e = sx2 + p2 - 2.f * d[r];
                colmin = fminf(colmin, e);
                nmin[k * 8 + r] = fminf(nmin[k * 8 + r], e);
            }
        }
        // Combine the two row-halves: full min over 16 s per column, per S-tile-set.
        colmin = fminf(colmin, __shfl_xor(colmin, 16, 32));
        if (!hi) pclsum += colmin * amp[(size_t)n * PP];
    }

    // Reduce nmin over the 16 columns (lanes) within each half-wave.
#pragma unroll
    for (int i = 0; i < STILES * 8; ++i) {
        float v = nmin[i];
        v = fminf(v, __shfl_xor(v, 1, 32));
        v = fminf(v, __shfl_xor(v, 2, 32));
        v = fminf(v, __shfl_xor(v, 4, 32));
        v = fminf(v, __shfl_xor(v, 8, 32));
        nmin[i] = v;
    }
    // lane 0 holds per-s min for s = 16k+r; lane 16 for s = 16k+8+r.
    if (col == 0) {
#pragma unroll
        for (int k = 0; k < STILES; ++k)
#pragma unroll
            for (int r = 0; r < 8; ++r)
                lds_prim[wave * SS + 16 * k + r + (hi ? 8 : 0)] = nmin[k * 8 + r];
    }

    // Block-reduce pclsum.
    lds_red[threadIdx.x] = pclsum;
    __syncthreads();
    for (int off = 128; off > 0; off >>= 1) {
        if ((int)threadIdx.x < off) lds_red[threadIdx.x] += lds_red[threadIdx.x + off];
        __syncthreads();
    }
    if (threadIdx.x == 0) pclpart[bp] = lds_red[0];
    __syncthreads();

    // prim2pcl: min over waves per s, then mean over s.
    float v = 0.f;
    if (threadIdx.x < SS) {
        float m = lds_prim[threadIdx.x];
#pragma unroll
        for (int w = 1; w < 8; ++w) m = fminf(m, lds_prim[w * SS + threadIdx.x]);
        v = m;
    }
    lds_red[threadIdx.x] = v;
    __syncthreads();
    for (int off = 128; off > 0; off >>= 1) {
        if ((int)threadIdx.x < off) lds_red[threadIdx.x] += lds_red[threadIdx.x + off];
        __syncthreads();
    }
    if (threadIdx.x == 0) prim2[bp] = lds_red[0] * (1.f / (float)SS);
}

// ---------------------------------------------------------------------------
// Kernel 3: cuboid projection loss. One thread per (b,n,p). Block partial sums.
// ---------------------------------------------------------------------------
__global__ __launch_bounds__(256)
void cuboid_kernel(const float* __restrict__ pc_inver,
                   const float* __restrict__ normals_inver,
                   const float* __restrict__ scale,
                   const float* __restrict__ am,
                   float* __restrict__ cubpart) {
    __shared__ float red[256];
    int gid = blockIdx.x * blockDim.x + threadIdx.x;  // over BB*NN*PP
    float val = 0.f;
    if (gid < BB * NN * PP) {
        int p = gid % PP;
        int n = (gid / PP) % NN;
        int b = gid / (PP * NN);
        size_t base = ((size_t)(b * PP + p) * NN + n) * 3;
        float cx = pc_inver[base + 0], cy = pc_inver[base + 1], cz = pc_inver[base + 2];
        float nx = normals_inver[base + 0], ny = normals_inver[base + 1], nz = normals_inver[base + 2];
        // argmax over 6 face cosines == argmax over (-nx, nx, -ny, ny, -nz, nz)
        float vals[6] = {-nx, nx, -ny, ny, -nz, nz};
        int idx = 0;
        float best = vals[0];
#pragma unroll
        for (int f = 1; f < 6; ++f) {
            if (vals[f] > best) { best = vals[f]; idx = f; }
        }
        float sx = scale[(b * PP + p) * 3 + 0];
        float sy = scale[(b * PP + p) * 3 + 1];
        float sz = scale[(b * PP + p) * 3 + 2];
        int axis = idx >> 1;
        float sgn = (idx & 1) ? 1.f : -1.f;
        float qx = (axis == 0) ? sgn * sx : fminf(fmaxf(cx, -sx), sx);
        float qy = (axis == 1) ? sgn * sy : fminf(fmaxf(cy, -sy), sy);
        float qz = (axis == 2) ? sgn * sz : fminf(fmaxf(cz, -sz), sz);
        float dx = qx - cx, dy = qy - cy, dz = qz - cz;
        float d = dx * dx + dy * dy + dz * dz;
        val = d * am[((size_t)b * NN + n) * PP + p];
    }
    red[threadIdx.x] = val;
    __syncthreads();
    for (int off = 128; off > 0; off >>= 1) {
        if ((int)threadIdx.x < off) red[threadIdx.x] += red[threadIdx.x + off];
        __syncthreads();
    }
    if (threadIdx.x == 0) cubpart[blockIdx.x] = red[0];
}

// ---------------------------------------------------------------------------
// Kernel 4: column sums of assign over N. One block per (b,p).
// ---------------------------------------------------------------------------
__global__ __launch_bounds__(256)
void colsum_kernel(const float* __restrict__ am, float* __restrict__ colsum) {
    __shared__ float red[256];
    int bp = blockIdx.x;
    int b = bp / PP;
    int p = bp % PP;
    const float* __restrict__ amp = am + (size_t)b * NN * PP + p;
    float s = 0.f;
    for (int n = threadIdx.x; n < NN; n += 256) s += amp[(size_t)n * PP];
    red[threadIdx.x] = s;
    __syncthreads();
    for (int off = 128; off > 0; off >>= 1) {
        if ((int)threadIdx.x < off) red[threadIdx.x] += red[threadIdx.x + off];
        __syncthreads();
    }
    if (threadIdx.x == 0) colsum[bp] = red[0];
}

// ---------------------------------------------------------------------------
// Kernel 5: combine everything into the scalar loss.
// ---------------------------------------------------------------------------
__global__ void finalize_kernel(const float* __restrict__ cubpart,
                                const float* __restrict__ pclpart,
                                const float* __restrict__ prim2,
                                const float* __restrict__ colsum,
                                const float* __restrict__ exist,
                                float* __restrict__ out) {
    if (threadIdx.x != 0 || blockIdx.x != 0) return;
    float cubsum = 0.f;
    for (int i = 0; i < CUB_BLOCKS; ++i) cubsum += cubpart[i];
    float cub = cubsum / (float)(BB * NN);

    float pclsum = 0.f;
    for (int i = 0; i < BP; ++i) pclsum += pclpart[i];
    float pcl = pclsum / (float)(BB * NN);

    float prim = 0.f;
    for (int b = 0; b < BB; ++b) {
        float sp = 0.f, se = 0.f;
        for (int p = 0; p < PP; ++p) {
            float e = exist[b * PP + p];
            sp += prim2[b * PP + p] * e;
            se += e;
        }
        prim += sp / (se + 1e-6f);
    }
    prim /= (float)BB;

    float bce = 0.f;
    for (int i = 0; i < BP; ++i) {
        float g = colsum[i] > 24.f ? 1.f : 0.f;
        float pe = exist[i];
        bce += g * logf(pe) + (1.f - g) * log1pf(-pe);
    }
    bce = -bce / (float)BP;

    float sps = 0.f;
    for (int b = 0; b < BB; ++b) {
        float m = 0.f;
        for (int p = 0; p < PP; ++p)
            m += sqrtf(colsum[b * PP + p] / (float)NN + 0.01f);
        m /= (float)PP;
        sps += m * m;
    }
    sps /= (float)BB;

    out[0] = 0.1f * cub + 1.0f * (pcl + prim) + 0.01f * bce + 0.01f * sps;
}

// ---------------------------------------------------------------------------
extern "C" void kernel_launch(void* const* d_in, const int* in_sizes, int n_in,
                              void* d_out, int out_size, void* d_ws, size_t ws_size,
                              hipStream_t stream) {
    const float* pc     = (const float*)d_in[0];
    const float* nrm    = (const float*)d_in[1];
    const float* scale  = (const float*)d_in[2];
    const float* shape  = (const float*)d_in[3];
    const float* exist  = (const float*)d_in[4];
    const float* am     = (const float*)d_in[5];
    const float* etas   = (const float*)d_in[6];
    const float* omegas = (const float*)d_in[7];
    (void)in_sizes; (void)n_in; (void)out_size; (void)ws_size;

    float* ws      = (float*)d_ws;
    float* X       = ws;                       // BP*SS*3 = 36864
    float* x2      = X + BP * SS * 3;          // BP*SS   = 12288
    float* pclpart = x2 + BP * SS;             // BP      = 96
    float* prim2   = pclpart + BP;             // BP      = 96
    float* cubpart = prim2 + BP;               // 1152
    float* colsum  = cubpart + CUB_BLOCKS;     // 96

    sample_kernel<<<(BP * SS) / 256, 256, 0, stream>>>(scale, shape, etas, omegas, X, x2);
    wmma_chamfer_kernel<<<BP, 256, 0, stream>>>(pc, am, X, x2, pclpart, prim2);
    cuboid_kernel<<<CUB_BLOCKS, 256, 0, stream>>>(pc, nrm, scale, am, cubpart);
    colsum_kernel<<<BP, 256, 0, stream>>>(am, colsum);
    finalize_kernel<<<1, 32, 0, stream>>>(cubpart, pclpart, prim2, colsum, exist,
                                          (float*)d_out);
}